// BUIR_NB_38053410242785
// MI455X (gfx1250) — compile-verified
//
#include <hip/hip_runtime.h>
#include <hip/hip_bf16.h>
#include <math.h>

typedef __attribute__((ext_vector_type(2))) float v2f;
typedef __attribute__((ext_vector_type(8))) float v8f;

#define LATENT 64
#define NEG_SLOPE 0.2f

// ---------------------------------------------------------------------------
// H[M x 64] = X[M x 64] @ W[64 x 64] with V_WMMA_F32_16X16X4_F32 (full f32).
// One wave per 16x16 output tile; 4 waves/block cover the 4 N-tiles.
// W staged in LDS. M is padded-safe via row clamping (EXEC stays all-ones
// at the WMMA; only the stores are predicated).
// ---------------------------------------------------------------------------
__global__ __launch_bounds__(128)
void gemm_xw_kernel(const float* __restrict__ X, const float* __restrict__ W,
                    float* __restrict__ H, int M) {
  __shared__ float sW[64 * 64];
  {
    const float4* s = (const float4*)W;
    float4* d = (float4*)sW;
    int t = threadIdx.x;
#pragma unroll
    for (int i = 0; i < 8; ++i) d[t + i * 128] = s[t + i * 128];
  }
  __syncthreads();

  const int lane  = threadIdx.x & 31;
  const int wv    = threadIdx.x >> 5;      // N tile 0..3
  const int m0    = blockIdx.x * 16;
  const int r     = lane & 15;
  const int khalf = (lane >> 4) << 1;      // 0 or 2 (ISA A-layout)
  const int c     = (lane & 15) + wv * 16;

  int row = m0 + r; if (row >= M) row = M - 1;           // clamp, keep EXEC full
  const float* xrow = X + (size_t)row * LATENT;

  v8f acc = {};
#pragma unroll
  for (int kk = 0; kk < 16; ++kk) {
    const int k = kk * 4;
    v2f a = *(const v2f*)&xrow[k + khalf];               // A[r][k+khalf .. +1]
    v2f b;
    b.x = sW[(k + khalf) * 64 + c];                      // B[k+khalf][c]
    b.y = sW[(k + khalf + 1) * 64 + c];
    acc = __builtin_amdgcn_wmma_f32_16x16x4_f32(false, a, false, b,
                                                (short)0, acc, false, false);
  }
  const int rbase = m0 + ((lane >> 4) << 3);             // C/D: M=i / M=i+8 halves
#pragma unroll
  for (int i = 0; i < 8; ++i)
    if (rbase + i < M) H[(size_t)(rbase + i) * LATENT + c] = acc[i];
}

// ---------------------------------------------------------------------------
// Per-node: hs = h . att_src, hd = h . att_dst (wave per node, shuffle reduce)
// Also resets m=-inf, den=0 and seeds xnext row with the layer bias so the
// edge scatter lands directly on segment_sum(msg) + b.
// ---------------------------------------------------------------------------
__global__ __launch_bounds__(256)
void att_init_kernel(const float* __restrict__ H,
                     const float* __restrict__ a_src, const float* __restrict__ a_dst,
                     const float* __restrict__ bias,
                     float* __restrict__ hs, float* __restrict__ hd,
                     float* __restrict__ m, float* __restrict__ den,
                     float* __restrict__ xnext, int nNodes) {
  int node = blockIdx.x * 8 + (threadIdx.x >> 5);
  if (node >= nNodes) return;
  int lane = threadIdx.x & 31;
  v2f h2 = *(const v2f*)&H[(size_t)node * LATENT + lane * 2];
  v2f as = *(const v2f*)&a_src[lane * 2];
  v2f ad = *(const v2f*)&a_dst[lane * 2];
  float ps = h2.x * as.x + h2.y * as.y;
  float pd = h2.x * ad.x + h2.y * ad.y;
#pragma unroll
  for (int off = 16; off > 0; off >>= 1) {
    ps += __shfl_down(ps, off, 32);
    pd += __shfl_down(pd, off, 32);
  }
  if (lane == 0) {
    hs[node] = ps;
    hd[node] = pd;
    m[node]  = __uint_as_float(0xff800000u);   // -inf
    den[node] = 0.0f;
  }
  *(v2f*)&xnext[(size_t)node * LATENT + lane * 2] = *(const v2f*)&bias[lane * 2];
}

// ---------------------------------------------------------------------------
// Edge pass 1: e = leaky_relu(hs[src]+hd[dst]); segment-max into m[dst]
// (int-punned float atomic max: atomicMax on int for >=0, atomicMin on uint
//  for <0; valid with -inf init).
// ---------------------------------------------------------------------------
__global__ void edge_max_kernel(const int* __restrict__ src, const int* __restrict__ dst,
                                int E, int nNodes,
                                const float* __restrict__ hs, const float* __restrict__ hd,
                                float* __restrict__ ebuf, float* __restrict__ m) {
  int j = blockIdx.x * blockDim.x + threadIdx.x;
  int total = E + nNodes;
  if (j >= total) return;
  int s, d;
  if (j < E) { s = src[j]; d = dst[j]; } else { s = j - E; d = s; }   // self loops
  float e = hs[s] + hd[d];
  e = (e > 0.0f) ? e : NEG_SLOPE * e;
  ebuf[j] = e;
  if (e >= 0.0f) atomicMax((int*)&m[d], __float_as_int(e));
  else           atomicMin((unsigned int*)&m[d], __float_as_uint(e));
}

// Edge pass 2: ex = exp(e - m[dst]); segment-sum into den[dst]
__global__ void edge_exp_kernel(const int* __restrict__ dst, int E, int nNodes,
                                const float* __restrict__ ebuf, const float* __restrict__ m,
                                float* __restrict__ exbuf, float* __restrict__ den) {
  int j = blockIdx.x * blockDim.x + threadIdx.x;
  int total = E + nNodes;
  if (j >= total) return;
  int d = (j < E) ? dst[j] : (j - E);
  float ex = expf(ebuf[j] - m[d]);
  exbuf[j] = ex;
  atomicAdd(&den[d], ex);
}

// Edge pass 3: alpha = ex/den[dst]; out[dst] += alpha * h[src]  (wave/edge)
__global__ __launch_bounds__(256)
void edge_scatter_kernel(const int* __restrict__ src, const int* __restrict__ dst,
                         int E, int nNodes,
                         const float* __restrict__ exbuf, const float* __restrict__ den,
                         const float* __restrict__ H, float* __restrict__ out) {
  int j = blockIdx.x * 8 + (threadIdx.x >> 5);
  int total = E + nNodes;
  if (j >= total) return;
  int lane = threadIdx.x & 31;
  int s, d;
  if (j < E) { s = src[j]; d = dst[j]; } else { s = j - E; d = s; }
  float alpha = exbuf[j] / den[d];
  v2f h2 = *(const v2f*)&H[(size_t)s * LATENT + lane * 2];
  float* o = out + (size_t)d * LATENT + lane * 2;
  atomicAdd(o,     alpha * h2.x);
  atomicAdd(o + 1, alpha * h2.y);
}

// ---------------------------------------------------------------------------
// Out[i][j] = sum_k Xall[off+idx[i]][k] * Wp[j][k] + bp[j]   (WMMA + gather)
// B = Wp^T, so b.x/b.y are contiguous in Wp's rows -> float2 LDS reads.
// ---------------------------------------------------------------------------
__global__ __launch_bounds__(128)
void pred_gemm_kernel(const float* __restrict__ Xall, const int* __restrict__ idx,
                      int nodeOff, const float* __restrict__ Wp,
                      const float* __restrict__ bp, float* __restrict__ Out, int M) {
  __shared__ float sW[64 * 64];
  __shared__ float sb[64];
  {
    const float4* s = (const float4*)Wp;
    float4* d = (float4*)sW;
    int t = threadIdx.x;
#pragma unroll
    for (int i = 0; i < 8; ++i) d[t + i * 128] = s[t + i * 128];
    if (t < 64) sb[t] = bp[t];
  }
  __syncthreads();

  const int lane  = threadIdx.x & 31;
  const int wv    = threadIdx.x >> 5;
  const int m0    = blockIdx.x * 16;
  const int r     = lane & 15;
  const int khalf = (lane >> 4) << 1;
  const int c     = (lane & 15) + wv * 16;

  int qi = m0 + r; if (qi >= M) qi = M - 1;
  const float* xrow = Xall + (size_t)(nodeOff + idx[qi]) * LATENT;

  v8f acc = {};
#pragma unroll
  for (int kk = 0; kk < 16; ++kk) {
    const int k = kk * 4;
    v2f a = *(const v2f*)&xrow[k + khalf];
    v2f b = *(const v2f*)&sW[c * 64 + (k + khalf)];      // Wp[c][k..k+1] = B[k..][c]
    acc = __builtin_amdgcn_wmma_f32_16x16x4_f32(false, a, false, b,
                                                (short)0, acc, false, false);
  }
  const int rbase = m0 + ((lane >> 4) << 3);
#pragma unroll
  for (int i = 0; i < 8; ++i)
    if (rbase + i < M) Out[(size_t)(rbase + i) * LATENT + c] = acc[i] + sb[c];
}

// Target gather: Out[i][:] = Xall[off+idx[i]][:]
__global__ void gather_kernel(const float* __restrict__ Xall, const int* __restrict__ idx,
                              int nodeOff, float* __restrict__ Out, int n) {
  int t = blockIdx.x * blockDim.x + threadIdx.x;
  if (t >= n * 32) return;
  int i = t >> 5, f = (t & 31) * 2;
  int node = nodeOff + idx[i];
  *(v2f*)&Out[(size_t)i * LATENT + f] = *(const v2f*)&Xall[(size_t)node * LATENT + f];
}

// ---------------------------------------------------------------------------
extern "C" void kernel_launch(void* const* d_in, const int* in_sizes, int n_in,
                              void* d_out, int out_size, void* d_ws, size_t ws_size,
                              hipStream_t stream) {
  const int*   user   = (const int*)d_in[0];
  const int*   item   = (const int*)d_in[1];
  const int*   edge   = (const int*)d_in[2];
  const float* uemb_o = (const float*)d_in[3];
  const float* iemb_o = (const float*)d_in[4];
  const float* W_o    = (const float*)d_in[5];
  const float* as_o   = (const float*)d_in[6];
  const float* ad_o   = (const float*)d_in[7];
  const float* b_o    = (const float*)d_in[8];
  const float* uemb_t = (const float*)d_in[9];
  const float* iemb_t = (const float*)d_in[10];
  const float* W_t    = (const float*)d_in[11];
  const float* as_t   = (const float*)d_in[12];
  const float* ad_t   = (const float*)d_in[13];
  const float* b_t    = (const float*)d_in[14];
  const float* predW  = (const float*)d_in[15];
  const float* predb  = (const float*)d_in[16];

  const int nq     = in_sizes[0];            // 4096
  const int E      = in_sizes[2] / 2;        // 2.4M
  const int users  = in_sizes[3] / LATENT;   // 100000
  const int items  = in_sizes[4] / LATENT;   // 50000
  const int nNodes = users + items;          // 150000 (multiple of 16)
  const int total  = E + nNodes;

  const int* srcIdx = edge;
  const int* dstIdx = edge + E;

  // workspace layout (floats)
  float* ws = (float*)d_ws;
  const size_t NF = (size_t)nNodes * LATENT;
  float* A     = ws;            // ping
  float* B     = A + NF;        // h
  float* C     = B + NF;        // pong / online final
  float* D     = C + NF;        // target final
  float* hs    = D + NF;
  float* hd    = hs + nNodes;
  float* mbuf  = hd + nNodes;
  float* den   = mbuf + nNodes;
  float* ebuf  = den + nNodes;
  float* exbuf = ebuf + total;

  const int gemmBlocks = (nNodes + 15) / 16;
  const int nodeWaves  = (nNodes + 7) / 8;
  const int edgeBlocks = (total + 255) / 256;
  const int edgeWaves  = (total + 7) / 8;

  auto run_layer = [&](const float* x, float* h, float* xn, int l,
                       const float* W, const float* as, const float* ad,
                       const float* bias) {
    gemm_xw_kernel<<<gemmBlocks, 128, 0, stream>>>(x, W + (size_t)l * 64 * 64, h, nNodes);
    att_init_kernel<<<nodeWaves, 256, 0, stream>>>(h, as + l * 64, ad + l * 64,
                                                   bias + l * 64, hs, hd, mbuf, den, xn, nNodes);
    edge_max_kernel<<<edgeBlocks, 256, 0, stream>>>(srcIdx, dstIdx, E, nNodes, hs, hd, ebuf, mbuf);
    edge_exp_kernel<<<edgeBlocks, 256, 0, stream>>>(dstIdx, E, nNodes, ebuf, mbuf, exbuf, den);
    edge_scatter_kernel<<<edgeWaves, 256, 0, stream>>>(srcIdx, dstIdx, E, nNodes, exbuf, den, h, xn);
  };

  // ---- online encoder: A -> C -> A -> C (final in C)
  hipMemcpyAsync(A, uemb_o, (size_t)users * LATENT * sizeof(float),
                 hipMemcpyDeviceToDevice, stream);
  hipMemcpyAsync(A + (size_t)users * LATENT, iemb_o,
                 (size_t)items * LATENT * sizeof(float), hipMemcpyDeviceToDevice, stream);
  run_layer(A, B, C, 0, W_o, as_o, ad_o, b_o);
  run_layer(C, B, A, 1, W_o, as_o, ad_o, b_o);
  run_layer(A, B, C, 2, W_o, as_o, ad_o, b_o);

  // ---- target encoder: A -> D -> A -> D (final in D; C untouched)
  hipMemcpyAsync(A, uemb_t, (size_t)users * LATENT * sizeof(float),
                 hipMemcpyDeviceToDevice, stream);
  hipMemcpyAsync(A + (size_t)users * LATENT, iemb_t,
                 (size_t)items * LATENT * sizeof(float), hipMemcpyDeviceToDevice, stream);
  run_layer(A, B, D, 0, W_t, as_t, ad_t, b_t);
  run_layer(D, B, A, 1, W_t, as_t, ad_t, b_t);
  run_layer(A, B, D, 2, W_t, as_t, ad_t, b_t);

  // ---- outputs: [u_pred | u_target | i_pred | i_target], each nq x 64
  float* out = (float*)d_out;
  const size_t seg = (size_t)nq * LATENT;
  const int predBlocks = (nq + 15) / 16;
  const int gatherBlocks = (nq * 32 + 255) / 256;

  pred_gemm_kernel<<<predBlocks, 128, 0, stream>>>(C, user, 0,     predW, predb, out, nq);
  gather_kernel<<<gatherBlocks, 256, 0, stream>>>(D, user, 0,     out + seg, nq);
  pred_gemm_kernel<<<predBlocks, 128, 0, stream>>>(C, item, users, predW, predb, out + 2 * seg, nq);
  gather_kernel<<<gatherBlocks, 256, 0, stream>>>(D, item, users, out + 3 * seg, nq);
}